// TGNMemory_11416023072996
// MI455X (gfx1250) — compile-verified
//
#include <hip/hip_runtime.h>
#include <math.h>

typedef float v2f __attribute__((ext_vector_type(2)));
typedef float v8f __attribute__((ext_vector_type(8)));

// ---------------------------------------------------------------------------
// K0: zero workspace (sums, cnt, assoc, lu_all)
// ---------------------------------------------------------------------------
__global__ void init_ws_kernel(float* __restrict__ sums, float* __restrict__ cnt,
                               int* __restrict__ assoc, int* __restrict__ lu,
                               long nsums, int N, int num_nodes) {
    long i = (long)blockIdx.x * blockDim.x + threadIdx.x;
    long stride = (long)gridDim.x * blockDim.x;
    for (long j = i; j < nsums; j += stride) sums[j] = 0.0f;
    for (long j = i; j < N; j += stride) cnt[j] = 0.0f;
    for (long j = i; j < num_nodes; j += stride) { assoc[j] = 0; lu[j] = 0; }
}

// ---------------------------------------------------------------------------
// K1: assoc[n_id[i]] = i
// ---------------------------------------------------------------------------
__global__ void build_assoc_kernel(const int* __restrict__ n_id,
                                   int* __restrict__ assoc, int N) {
    int i = blockIdx.x * blockDim.x + threadIdx.x;
    if (i < N) assoc[n_id[i]] = i;
}

// ---------------------------------------------------------------------------
// K2: per-edge message construction + segment-sum scatter + ts scatter-max.
// One 128-thread block per edge; threads stride across D_IN features.
// ---------------------------------------------------------------------------
__global__ void edge_scatter_kernel(const int* __restrict__ src, const int* __restrict__ dst,
                                    const int* __restrict__ t, const float* __restrict__ raw_msg,
                                    const float* __restrict__ memory, const int* __restrict__ last_update,
                                    const float* __restrict__ time_w, const float* __restrict__ time_b,
                                    const int* __restrict__ assoc,
                                    float* __restrict__ sums, float* __restrict__ cnt,
                                    int* __restrict__ lu_all,
                                    int E, int H, int RAW, int TD, int D_IN) {
    int e = blockIdx.x;
    if (e >= E) return;
    int s  = src[e];
    int d0 = dst[e];
    int te = t[e];
    int idx = assoc[s];
    float t_rel = (float)(te - last_update[s]);
    if (threadIdx.x == 0) {
        atomicAdd(&cnt[idx], 1.0f);
        atomicMax(&lu_all[s], te);
    }
    float* srow = sums + (size_t)idx * D_IN;
    const float* msrc = memory + (size_t)s  * H;
    const float* mdst = memory + (size_t)d0 * H;
    const float* rmsg = raw_msg + (size_t)e * RAW;
    int off_raw = 2 * H;
    int off_te  = 2 * H + RAW;
    for (int f = threadIdx.x; f < D_IN; f += blockDim.x) {
        float v;
        if (f < H)             v = msrc[f];
        else if (f < 2 * H)    v = mdst[f - H];
        else if (f < off_te)   v = rmsg[f - off_raw];
        else                   v = cosf(t_rel * time_w[f - off_te] + time_b[f - off_te]);
        atomicAdd(&srow[f], v);
    }
}

// ---------------------------------------------------------------------------
// K3: aggr = sums / max(cnt, 1)   (in place)
// ---------------------------------------------------------------------------
__global__ void divide_kernel(float* __restrict__ sums, const float* __restrict__ cnt,
                              long total, int D_IN) {
    long i = (long)blockIdx.x * blockDim.x + threadIdx.x;
    long stride = (long)gridDim.x * blockDim.x;
    for (long j = i; j < total; j += stride) {
        float c = cnt[j / D_IN];
        sums[j] = sums[j] / fmaxf(c, 1.0f);
    }
}

__device__ __forceinline__ float sigmoidf_dev(float x) {
    return 1.0f / (1.0f + expf(-x));
}

// ---------------------------------------------------------------------------
// K4: fused GRU cell via fp32 WMMA (V_WMMA_F32_16X16X4_F32).
// One wave per 16x16 output tile of new_memory. Six accumulator chains:
// gi_{r,z,n} over K=D_IN against W_ih, gh_{r,z,n} over K=H against W_hh.
// Column indices clamped (not masked) so EXEC is all-ones through WMMA.
// ---------------------------------------------------------------------------
__global__ void gru_wmma_kernel(const float* __restrict__ aggr,
                                const float* __restrict__ memory,
                                const int* __restrict__ n_id,
                                const float* __restrict__ W_ih,
                                const float* __restrict__ W_hh,
                                const float* __restrict__ b_ih,
                                const float* __restrict__ b_hh,
                                float* __restrict__ out,
                                int N, int H, int D_IN,
                                int tilesTotal, int colTiles) {
    int lane = threadIdx.x & 31;
    int wave = threadIdx.x >> 5;
    int tile = blockIdx.x * (blockDim.x >> 5) + wave;
    if (tile >= tilesTotal) return;  // wave-uniform exit

    int rowTile = tile / colTiles;
    int colTile = tile - rowTile * colTiles;
    int lrow = lane & 15;
    int half = lane >> 4;

    int arow  = rowTile * 16 + lrow;
    int arowC = arow < N ? arow : N - 1;          // clamped A row
    int bcol  = colTile * 16 + lrow;
    int bcolC = bcol < H ? bcol : H - 1;          // clamped B column

    v8f accR = {}, accZ = {}, accN = {};          // gi accumulators
    v8f hccR = {}, hccZ = {}, hccN = {};          // gh accumulators

    // ---- gi = aggr @ W_ih^T ----
    const float* ap = aggr + (size_t)arowC * D_IN;
    const float* wr = W_ih + (size_t)(0 * H + bcolC) * D_IN;
    const float* wz = W_ih + (size_t)(1 * H + bcolC) * D_IN;
    const float* wn = W_ih + (size_t)(2 * H + bcolC) * D_IN;
    for (int k = 0; k < D_IN; k += 4) {
        int kl = k + 2 * half;
        v2f a;  a.x  = ap[kl]; a.y  = ap[kl + 1];
        v2f br; br.x = wr[kl]; br.y = wr[kl + 1];
        v2f bz; bz.x = wz[kl]; bz.y = wz[kl + 1];
        v2f bn; bn.x = wn[kl]; bn.y = wn[kl + 1];
        accR = __builtin_amdgcn_wmma_f32_16x16x4_f32(false, a, false, br, (short)0, accR, false, false);
        accZ = __builtin_amdgcn_wmma_f32_16x16x4_f32(false, a, false, bz, (short)0, accZ, false, false);
        accN = __builtin_amdgcn_wmma_f32_16x16x4_f32(false, a, false, bn, (short)0, accN, false, false);
    }

    // ---- gh = h @ W_hh^T  with h = memory[n_id[row]] ----
    int hrow = n_id[arowC];
    const float* hp = memory + (size_t)hrow * H;
    const float* vr = W_hh + (size_t)(0 * H + bcolC) * H;
    const float* vz = W_hh + (size_t)(1 * H + bcolC) * H;
    const float* vn = W_hh + (size_t)(2 * H + bcolC) * H;
    for (int k = 0; k < H; k += 4) {
        int kl = k + 2 * half;
        v2f a;  a.x  = hp[kl]; a.y  = hp[kl + 1];
        v2f br; br.x = vr[kl]; br.y = vr[kl + 1];
        v2f bz; bz.x = vz[kl]; bz.y = vz[kl + 1];
        v2f bn; bn.x = vn[kl]; bn.y = vn[kl + 1];
        hccR = __builtin_amdgcn_wmma_f32_16x16x4_f32(false, a, false, br, (short)0, hccR, false, false);
        hccZ = __builtin_amdgcn_wmma_f32_16x16x4_f32(false, a, false, bz, (short)0, hccZ, false, false);
        hccN = __builtin_amdgcn_wmma_f32_16x16x4_f32(false, a, false, bn, (short)0, hccN, false, false);
    }

    // ---- gates + store (D layout: VGPR r -> row r + 8*half, col = lrow) ----
    int gcol = colTile * 16 + lrow;
    #pragma unroll
    for (int r = 0; r < 8; r++) {
        int m = r + 8 * half;
        int grow = rowTile * 16 + m;
        if (grow < N && gcol < H) {
            float gir = accR[r] + b_ih[0 * H + gcol];
            float giz = accZ[r] + b_ih[1 * H + gcol];
            float gin = accN[r] + b_ih[2 * H + gcol];
            float ghr = hccR[r] + b_hh[0 * H + gcol];
            float ghz = hccZ[r] + b_hh[1 * H + gcol];
            float ghn = hccN[r] + b_hh[2 * H + gcol];
            float rg = sigmoidf_dev(gir + ghr);
            float zg = sigmoidf_dev(giz + ghz);
            float ng = tanhf(gin + rg * ghn);
            float hv = memory[(size_t)n_id[grow] * H + gcol];
            out[(size_t)grow * H + gcol] = (1.0f - zg) * ng + zg * hv;
        }
    }
}

// ---------------------------------------------------------------------------
// K5: new_last_update = max(lu_all, 0)[n_id], emitted as float
// ---------------------------------------------------------------------------
__global__ void lu_out_kernel(const int* __restrict__ n_id, const int* __restrict__ lu_all,
                              float* __restrict__ out_lu, int N) {
    int i = blockIdx.x * blockDim.x + threadIdx.x;
    if (i < N) {
        int v = lu_all[n_id[i]];
        out_lu[i] = (float)(v > 0 ? v : 0);
    }
}

extern "C" void kernel_launch(void* const* d_in, const int* in_sizes, int n_in,
                              void* d_out, int out_size, void* d_ws, size_t ws_size,
                              hipStream_t stream) {
    const int*   n_id        = (const int*)d_in[0];
    const int*   src         = (const int*)d_in[1];
    const int*   dst         = (const int*)d_in[2];
    const int*   t           = (const int*)d_in[3];
    const float* raw_msg     = (const float*)d_in[4];
    const float* memory      = (const float*)d_in[5];
    const int*   last_update = (const int*)d_in[6];
    const float* time_w      = (const float*)d_in[7];
    const float* time_b      = (const float*)d_in[8];
    const float* W_ih        = (const float*)d_in[9];
    const float* W_hh        = (const float*)d_in[10];
    const float* b_ih        = (const float*)d_in[11];
    const float* b_hh        = (const float*)d_in[12];

    const int N         = in_sizes[0];
    const int E         = in_sizes[1];
    const int num_nodes = in_sizes[6];
    const int TD        = in_sizes[7];
    const int H         = in_sizes[5] / num_nodes;     // 100
    const int RAW       = in_sizes[4] / E;             // 172
    const int D_IN      = 2 * H + RAW + TD;            // 472

    // Workspace layout
    char* ws = (char*)d_ws;
    float* sums = (float*)ws;                                      // N * D_IN floats
    size_t off = (size_t)N * D_IN * sizeof(float);
    float* cnt  = (float*)(ws + off);  off += (size_t)N * sizeof(float);
    int*   assoc = (int*)(ws + off);   off += (size_t)num_nodes * sizeof(int);
    int*   lu_all = (int*)(ws + off);  off += (size_t)num_nodes * sizeof(int);
    (void)ws_size;

    float* out_mem = (float*)d_out;
    float* out_lu  = out_mem + (size_t)N * H;
    (void)out_size;

    const long nsums = (long)N * D_IN;

    init_ws_kernel<<<4096, 256, 0, stream>>>(sums, cnt, assoc, lu_all, nsums, N, num_nodes);
    build_assoc_kernel<<<(N + 255) / 256, 256, 0, stream>>>(n_id, assoc, N);
    edge_scatter_kernel<<<E, 128, 0, stream>>>(src, dst, t, raw_msg, memory, last_update,
                                               time_w, time_b, assoc,
                                               sums, cnt, lu_all, E, H, RAW, TD, D_IN);
    divide_kernel<<<4096, 256, 0, stream>>>(sums, cnt, nsums, D_IN);

    const int rowTiles = (N + 15) / 16;
    const int colTiles = (H + 15) / 16;
    const int tilesTotal = rowTiles * colTiles;
    const int wavesPerBlock = 8;   // 256 threads
    gru_wmma_kernel<<<(tilesTotal + wavesPerBlock - 1) / wavesPerBlock, 32 * wavesPerBlock, 0, stream>>>(
        sums, memory, n_id, W_ih, W_hh, b_ih, b_hh, out_mem, N, H, D_IN, tilesTotal, colTiles);

    lu_out_kernel<<<(N + 255) / 256, 256, 0, stream>>>(n_id, lu_all, out_lu, N);
}